// TAG_76776835383355
// MI455X (gfx1250) — compile-verified
//
#include <hip/hip_runtime.h>
#include <hip/hip_bf16.h>

typedef __attribute__((ext_vector_type(2))) float v2f;
typedef __attribute__((ext_vector_type(8))) float v8f;

#define NNODES 100000
#define NEDGES 1600000
#define NGRAPH 128

// ---------------- utility ----------------
__global__ void zero_kernel(float* __restrict__ p, int n) {
    int i = blockIdx.x * blockDim.x + threadIdx.x;
    if (i < n) p[i] = 0.0f;
}

// ---------------- gcn_norm ----------------
__global__ void deg_kernel(const int* __restrict__ dst, const float* __restrict__ w,
                           float* __restrict__ deg, int nE) {
    int e = blockIdx.x * blockDim.x + threadIdx.x;
    if (e < nE) atomicAdd(&deg[dst[e]], w[e]);
}

__global__ void dinv_kernel(const float* __restrict__ deg, float* __restrict__ dinv, int n) {
    int i = blockIdx.x * blockDim.x + threadIdx.x;
    if (i < n) {
        float d = deg[i];
        dinv[i] = (d > 0.0f) ? rsqrtf(fmaxf(d, 1e-30f)) : 0.0f;
    }
}

__global__ void norm_kernel(const int* __restrict__ src, const int* __restrict__ dst,
                            const float* __restrict__ w, const float* __restrict__ dinv,
                            float* __restrict__ norm, int nE) {
    int e = blockIdx.x * blockDim.x + threadIdx.x;
    if (e < nE) norm[e] = dinv[src[e]] * w[e] * dinv[dst[e]];
}

// ---------------- sparse propagate: hout[dst] += hin[src]*norm ----------------
// F lanes per edge (F = 32 or 8); feature reads coalesced across lanes.
template <int F>
__global__ void prop_kernel(const float* __restrict__ hin, float* __restrict__ hout,
                            const int* __restrict__ src, const int* __restrict__ dst,
                            const float* __restrict__ norm, int nE) {
    int tid = blockIdx.x * blockDim.x + threadIdx.x;
    int e = tid / F;
    int f = tid % F;
    if (e >= nE) return;
    int s = src[e];
    int d = dst[e];
    float v = hin[s * F + f] * norm[e];
    atomicAdd(&hout[d * F + f], v);
}

// ---------------- WMMA matmul over concatenated hop features ----------------
// out[n, 0:NOUT] = relu( sum_k [h0|h1|h2|h3][n,k] * W[k, 0:NOUT] + bias )
// One wave per 16-node tile; K = 4*FE, stepped by 4 via V_WMMA_F32_16X16X4_F32.
// Weights staged in LDS, zero-padded to 16 columns -> branch-free ds_load B reads.
template <int FE, int NOUT>
__global__ void tag_matmul_kernel(const float* __restrict__ h0, const float* __restrict__ h1,
                                  const float* __restrict__ h2, const float* __restrict__ h3,
                                  const float* __restrict__ W, const float* __restrict__ bias,
                                  float* __restrict__ out, int nNodes) {
    constexpr int K = 4 * FE;
    __shared__ float sW[K * 16];

    // Cooperative LDS fill (whole block, before any wave exits)
    for (int idx = threadIdx.x; idx < K * 16; idx += 256) {
        int k = idx >> 4;
        int n = idx & 15;
        sW[idx] = (n < NOUT) ? W[k * NOUT + n] : 0.0f;
    }
    __syncthreads();

    int wave = (blockIdx.x * blockDim.x + threadIdx.x) >> 5;
    int lane = threadIdx.x & 31;
    int row0 = wave * 16;
    if (row0 >= nNodes) return;  // wave-uniform: EXEC stays all-ones for WMMA

    const int M    = lane & 15;              // A-fragment row
    const int klo  = (lane >> 4) << 1;       // K sub-offset (0 for lanes 0-15, 2 for 16-31)
    const int ncol = lane & 15;              // B/D column
    int nodeA = row0 + M;
    if (nodeA > nNodes - 1) nodeA = nNodes - 1;

    v8f c = {};
#pragma unroll
    for (int kb = 0; kb < K; kb += 4) {
        const int hop = kb / FE;  // compile-time after unroll; k0,k1 stay in same hop block
        const float* __restrict__ ha = (hop == 0) ? h0 : (hop == 1) ? h1 : (hop == 2) ? h2 : h3;
        int k0 = kb + klo;
        int i0 = k0 % FE;
        v2f a, b;
        a.x = ha[nodeA * FE + i0];
        a.y = ha[nodeA * FE + i0 + 1];
        b.x = sW[k0 * 16 + ncol];
        b.y = sW[(k0 + 1) * 16 + ncol];
        c = __builtin_amdgcn_wmma_f32_16x16x4_f32(false, a, false, b, (short)0, c, false, false);
    }

    if (ncol < NOUT) {
        float bv = bias[ncol];
        int rbase = row0 + ((lane >> 4) << 3);  // lanes 0-15: rows M=0..7; 16-31: M=8..15
#pragma unroll
        for (int r = 0; r < 8; ++r) {
            int row = rbase + r;
            if (row < nNodes) {
                float v = c[r] + bv;
                out[row * NOUT + ncol] = (v > 0.0f) ? v : 0.0f;
            }
        }
    }
}

// ---------------- pooling: per-graph max + sum + count ----------------
// ReLU outputs are >= 0, so float max == int-bit max with 0 init.
__global__ void pool_kernel(const float* __restrict__ h, const int* __restrict__ batch,
                            float* __restrict__ mx, float* __restrict__ sm,
                            float* __restrict__ cnt, int n) {
    int tid = blockIdx.x * blockDim.x + threadIdx.x;
    int node = tid >> 4;
    int f = tid & 15;
    if (node >= n) return;
    int g = batch[node];
    float v = h[node * 16 + f];
    atomicMax((int*)&mx[g * 16 + f], __float_as_int(v));
    atomicAdd(&sm[g * 16 + f], v);
    if (f == 0) atomicAdd(&cnt[g], 1.0f);
}

// ---------------- final FC: concat(max, mean) (128x32) @ Wfc (32x2) + bfc ----------------
__global__ void fc_kernel(const float* __restrict__ mx, const float* __restrict__ sm,
                          const float* __restrict__ cnt, const float* __restrict__ Wfc,
                          const float* __restrict__ bfc, float* __restrict__ out) {
    int tid = blockIdx.x * blockDim.x + threadIdx.x;
    if (tid >= NGRAPH * 2) return;
    int g = tid >> 1;
    int o = tid & 1;
    float inv = 1.0f / fmaxf(cnt[g], 1.0f);
    float acc = bfc[o];
#pragma unroll
    for (int f = 0; f < 16; ++f) acc += mx[g * 16 + f] * Wfc[f * 2 + o];
#pragma unroll
    for (int f = 0; f < 16; ++f) acc += sm[g * 16 + f] * inv * Wfc[(16 + f) * 2 + o];
    out[g * 2 + o] = acc;
}

// ---------------- host launcher ----------------
extern "C" void kernel_launch(void* const* d_in, const int* in_sizes, int n_in,
                              void* d_out, int out_size, void* d_ws, size_t ws_size,
                              hipStream_t stream) {
    const float* x      = (const float*)d_in[0];   // (N, 32)
    const int*   eidx   = (const int*)d_in[1];     // (2, E)
    const int*   batch  = (const int*)d_in[2];     // (N,)
    const float* ew     = (const float*)d_in[3];   // (E,)
    const float* W1     = (const float*)d_in[4];   // (4,32,8)  -> flat (128,8)
    const float* b1     = (const float*)d_in[5];   // (8,)
    const float* W2     = (const float*)d_in[6];   // (4,8,16)  -> flat (32,16)
    const float* b2     = (const float*)d_in[7];   // (16,)
    const float* Wfc    = (const float*)d_in[8];   // (32,2)
    const float* bfc    = (const float*)d_in[9];   // (2,)
    float* out = (float*)d_out;

    const int* src = eidx;
    const int* dst = eidx + NEDGES;

    // ---- workspace layout (floats) ----
    float* ws = (float*)d_ws;
    size_t off = 0;
    float* deg  = ws + off; off += NNODES;
    float* dinv = ws + off; off += NNODES;
    float* nrm  = ws + off; off += NEDGES;
    float* h1   = ws + off; off += (size_t)NNODES * 32;
    float* h2   = ws + off; off += (size_t)NNODES * 32;
    float* h3   = ws + off; off += (size_t)NNODES * 32;
    float* y1   = ws + off; off += (size_t)NNODES * 8;   // layer-1 output == g0
    float* g1   = ws + off; off += (size_t)NNODES * 8;
    float* g2   = ws + off; off += (size_t)NNODES * 8;
    float* g3   = ws + off; off += (size_t)NNODES * 8;
    float* y2   = ws + off; off += (size_t)NNODES * 16;  // layer-2 output
    float* mx   = ws + off; off += NGRAPH * 16;
    float* sm   = ws + off; off += NGRAPH * 16;
    float* cnt  = ws + off; off += NGRAPH;

    const int BT = 256;
    auto blocks = [](long long n, int bt) { return (int)((n + bt - 1) / bt); };

    // gcn_norm
    zero_kernel<<<blocks(NNODES, BT), BT, 0, stream>>>(deg, NNODES);
    deg_kernel<<<blocks(NEDGES, BT), BT, 0, stream>>>(dst, ew, deg, NEDGES);
    dinv_kernel<<<blocks(NNODES, BT), BT, 0, stream>>>(deg, dinv, NNODES);
    norm_kernel<<<blocks(NEDGES, BT), BT, 0, stream>>>(src, dst, ew, dinv, nrm, NEDGES);

    // layer 1 hops (32 features)
    zero_kernel<<<blocks((long long)NNODES * 32, BT), BT, 0, stream>>>(h1, NNODES * 32);
    prop_kernel<32><<<blocks((long long)NEDGES * 32, BT), BT, 0, stream>>>(x, h1, src, dst, nrm, NEDGES);
    zero_kernel<<<blocks((long long)NNODES * 32, BT), BT, 0, stream>>>(h2, NNODES * 32);
    prop_kernel<32><<<blocks((long long)NEDGES * 32, BT), BT, 0, stream>>>(h1, h2, src, dst, nrm, NEDGES);
    zero_kernel<<<blocks((long long)NNODES * 32, BT), BT, 0, stream>>>(h3, NNODES * 32);
    prop_kernel<32><<<blocks((long long)NEDGES * 32, BT), BT, 0, stream>>>(h2, h3, src, dst, nrm, NEDGES);

    // layer 1 matmul: (N x 128) @ (128 x 8), relu  -- WMMA f32 16x16x4
    {
        int waves = (NNODES + 15) / 16;
        tag_matmul_kernel<32, 8><<<blocks((long long)waves * 32, BT), BT, 0, stream>>>(
            x, h1, h2, h3, W1, b1, y1, NNODES);
    }

    // layer 2 hops (8 features)
    zero_kernel<<<blocks((long long)NNODES * 8, BT), BT, 0, stream>>>(g1, NNODES * 8);
    prop_kernel<8><<<blocks((long long)NEDGES * 8, BT), BT, 0, stream>>>(y1, g1, src, dst, nrm, NEDGES);
    zero_kernel<<<blocks((long long)NNODES * 8, BT), BT, 0, stream>>>(g2, NNODES * 8);
    prop_kernel<8><<<blocks((long long)NEDGES * 8, BT), BT, 0, stream>>>(g1, g2, src, dst, nrm, NEDGES);
    zero_kernel<<<blocks((long long)NNODES * 8, BT), BT, 0, stream>>>(g3, NNODES * 8);
    prop_kernel<8><<<blocks((long long)NEDGES * 8, BT), BT, 0, stream>>>(g2, g3, src, dst, nrm, NEDGES);

    // layer 2 matmul: (N x 32) @ (32 x 16), relu  -- WMMA f32 16x16x4
    {
        int waves = (NNODES + 15) / 16;
        tag_matmul_kernel<8, 16><<<blocks((long long)waves * 32, BT), BT, 0, stream>>>(
            y1, g1, g2, g3, W2, b2, y2, NNODES);
    }

    // pooling
    zero_kernel<<<blocks(NGRAPH * 16 * 2 + NGRAPH, BT), BT, 0, stream>>>(mx, NGRAPH * 16 * 2 + NGRAPH);
    pool_kernel<<<blocks((long long)NNODES * 16, BT), BT, 0, stream>>>(y2, batch, mx, sm, cnt, NNODES);

    // final FC -> (128, 2)
    fc_kernel<<<1, 256, 0, stream>>>(mx, sm, cnt, Wfc, bfc, out);
}